// MetaGatedTitansLayer_10840497455730
// MI455X (gfx1250) — compile-verified
//
#include <hip/hip_runtime.h>
#include <cmath>

#define B_ 256
#define D_ 512

typedef float v2f __attribute__((ext_vector_type(2)));
typedef float v8f __attribute__((ext_vector_type(8)));

// ---------------------------------------------------------------------------
// wave32 reduction
// ---------------------------------------------------------------------------
__device__ __forceinline__ float wave_reduce_sum(float v) {
#pragma unroll
  for (int off = 16; off > 0; off >>= 1)
    v += __shfl_xor(v, off, 32);
  return v;
}

// ---------------------------------------------------------------------------
// Row LayerNorm (optionally fused ReLU).  One 256-thread block per row, D=512.
// Safe in-place (each thread reads its elements into regs before writing).
// ---------------------------------------------------------------------------
__global__ void ln_rows(const float* __restrict__ x, const float* __restrict__ g,
                        const float* __restrict__ bb, float* __restrict__ y,
                        int relu) {
  const int row = blockIdx.x;
  const int t = threadIdx.x;
  const float* xr = x + (size_t)row * D_;
  float x0 = xr[t], x1 = xr[t + 256];
  __shared__ float s1[256], s2[256];
  s1[t] = x0 + x1;
  s2[t] = x0 * x0 + x1 * x1;
  __syncthreads();
#pragma unroll
  for (int off = 128; off > 0; off >>= 1) {
    if (t < off) { s1[t] += s1[t + off]; s2[t] += s2[t + off]; }
    __syncthreads();
  }
  const float mean = s1[0] * (1.0f / D_);
  const float var  = s2[0] * (1.0f / D_) - mean * mean;
  const float rs = rsqrtf(var + 1e-5f);
  float y0 = (x0 - mean) * rs * g[t]       + bb[t];
  float y1 = (x1 - mean) * rs * g[t + 256] + bb[t + 256];
  if (relu) { y0 = fmaxf(y0, 0.f); y1 = fmaxf(y1, 0.f); }
  float* yr = y + (size_t)row * D_;
  yr[t] = y0;
  yr[t + 256] = y1;
}

// ---------------------------------------------------------------------------
// In-place row L2 normalization.
// ---------------------------------------------------------------------------
__global__ void l2norm_rows(float* __restrict__ y) {
  const int row = blockIdx.x;
  const int t = threadIdx.x;
  float* yr = y + (size_t)row * D_;
  float y0 = yr[t], y1 = yr[t + 256];
  __shared__ float s[256];
  s[t] = y0 * y0 + y1 * y1;
  __syncthreads();
#pragma unroll
  for (int off = 128; off > 0; off >>= 1) {
    if (t < off) s[t] += s[t + off];
    __syncthreads();
  }
  const float inv = 1.0f / fmaxf(sqrtf(s[0]), 1e-12f);
  yr[t] = y0 * inv;
  yr[t + 256] = y1 * inv;
}

// ---------------------------------------------------------------------------
// WMMA fp32 GEMM (fast path, N multiple of 16):
//   C[M,N] = A[M,K] @ W[N,K]^T (+ bias[N])
// One wave32 per 16x16 tile; K-loop of V_WMMA_F32_16X16X4_F32.
// A (16x4 f32): lanes 0-15 hold M rows with K={k,k+1}; lanes 16-31 K={k+2,k+3}.
// B (4x16) mirrored from W rows.  C/D: 8 VGPRs, M=v / v+8 by lane half.
// No masking in the K loop -> pure load + wmma stream.
// ---------------------------------------------------------------------------
__global__ void gemm_wmma_f32(const float* __restrict__ A, const float* __restrict__ W,
                              const float* __restrict__ bias, float* __restrict__ C,
                              int K, int ldc, int ntn, int ntiles) {
  const int wave = (int)((blockIdx.x * blockDim.x + threadIdx.x) >> 5);
  if (wave >= ntiles) return;              // wave-uniform
  const int lane = threadIdx.x & 31;
  const int tm = wave / ntn;
  const int tn = wave - tm * ntn;
  const int m0 = tm << 4;
  const int n0 = tn << 4;
  const int half = lane >> 4;
  const int l16 = lane & 15;

  const float* Ap = A + (size_t)(m0 + l16) * K + (half << 1);
  const float* Wp = W + (size_t)(n0 + l16) * K + (half << 1);

  v8f acc = {};
#pragma unroll 4
  for (int kk = 0; kk < K; kk += 4) {
    v2f a = *(const v2f*)(Ap + kk);
    v2f b = *(const v2f*)(Wp + kk);
    acc = __builtin_amdgcn_wmma_f32_16x16x4_f32(
        false, a, false, b, (short)0, acc, false, false);
  }

  const float bv = bias ? bias[n0 + l16] : 0.0f;
#pragma unroll
  for (int v = 0; v < 8; ++v) {
    const int row = m0 + v + (half << 3);
    C[(size_t)row * ldc + (n0 + l16)] = acc[v] + bv;
  }
}

// ---------------------------------------------------------------------------
// WMMA fp32 GEMM with N-edge guard (for N=1026).  Value-masked W loads keep
// EXEC all-ones through every WMMA; divergent store guard only after K loop.
// ---------------------------------------------------------------------------
__global__ void gemm_wmma_f32_pad(const float* __restrict__ A, const float* __restrict__ W,
                                  const float* __restrict__ bias, float* __restrict__ C,
                                  int N, int K, int ldc, int ntn, int ntiles) {
  const int wave = (int)((blockIdx.x * blockDim.x + threadIdx.x) >> 5);
  if (wave >= ntiles) return;
  const int lane = threadIdx.x & 31;
  const int tm = wave / ntn;
  const int tn = wave - tm * ntn;
  const int m0 = tm << 4;
  const int n0 = tn << 4;
  const int half = lane >> 4;
  const int l16 = lane & 15;

  const int bcol = n0 + l16;
  const int bcolc = bcol < N ? bcol : N - 1;
  const float bmask = (bcol < N) ? 1.0f : 0.0f;   // v_cndmask, no EXEC change

  const float* Ap = A + (size_t)(m0 + l16) * K + (half << 1);
  const float* Wp = W + (size_t)bcolc * K + (half << 1);

  v8f acc = {};
#pragma unroll 4
  for (int kk = 0; kk < K; kk += 4) {
    v2f a = *(const v2f*)(Ap + kk);
    v2f b = *(const v2f*)(Wp + kk);
    b.x *= bmask;
    b.y *= bmask;
    acc = __builtin_amdgcn_wmma_f32_16x16x4_f32(
        false, a, false, b, (short)0, acc, false, false);
  }

  if (bcol < N) {                          // divergence only after all WMMAs
    const float bv = bias ? bias[bcol] : 0.0f;
#pragma unroll
    for (int v = 0; v < 8; ++v) {
      const int row = m0 + v + (half << 3);
      C[(size_t)row * ldc + bcol] = acc[v] + bv;
    }
  }
}

// ---------------------------------------------------------------------------
// memory_context[b,i] = dot(old_state[b,i,:], vec[b,:])
// One wave per state row; lane l owns elements {4l..4l+3} of each 128-chunk
// (coalesced b128 loads).
// ---------------------------------------------------------------------------
__global__ void state_matvec(const float* __restrict__ S, const float* __restrict__ vec,
                             float* __restrict__ out) {
  const int b = blockIdx.x;
  const int wave = threadIdx.x >> 5;
  const int lane = threadIdx.x & 31;
  const int i = blockIdx.y * 8 + wave;
  const float* row = S + ((size_t)b * D_ + i) * D_;
  const float* vv = vec + (size_t)b * D_;
  float partial = 0.f;
#pragma unroll
  for (int c = 0; c < 4; ++c) {
    const int j = c * 128 + lane * 4;
    const float4 s4 = *(const float4*)(row + j);
    const float4 v4 = *(const float4*)(vv + j);
    partial += s4.x * v4.x + s4.y * v4.y + s4.z * v4.z + s4.w * v4.w;
  }
  partial = wave_reduce_sum(partial);
  if (lane == 0) out[(size_t)b * D_ + i] = partial;
}

// ---------------------------------------------------------------------------
// Fused final pass: state row stays register-resident.
//   pred_i = dot(S[b,i,:], k[b,:]);  err_i = v[b,i] - pred_i
//   out    = (1-alpha_b) * S + (eta_b * err_i) * k
// One read + one write of the 268 MB state.
// ---------------------------------------------------------------------------
__global__ void state_update(const float* __restrict__ S, const float* __restrict__ kv,
                             const float* __restrict__ vv, const float* __restrict__ alpha,
                             const float* __restrict__ eta, float* __restrict__ out) {
  const int b = blockIdx.x;
  const int wave = threadIdx.x >> 5;
  const int lane = threadIdx.x & 31;
  const int i = blockIdx.y * 8 + wave;
  const float* row = S + ((size_t)b * D_ + i) * D_;
  const float* kb = kv + (size_t)b * D_;

  float4 s[4], k4[4];
  float partial = 0.f;
#pragma unroll
  for (int c = 0; c < 4; ++c) {
    const int j = c * 128 + lane * 4;
    s[c]  = *(const float4*)(row + j);
    k4[c] = *(const float4*)(kb + j);
    partial += s[c].x * k4[c].x + s[c].y * k4[c].y + s[c].z * k4[c].z + s[c].w * k4[c].w;
  }
  const float pred = wave_reduce_sum(partial);
  const float err = vv[(size_t)b * D_ + i] - pred;
  const float oma = 1.0f - alpha[b];
  const float ee = eta[b] * err;
  float* orow = out + ((size_t)b * D_ + i) * D_;
#pragma unroll
  for (int c = 0; c < 4; ++c) {
    float4 r;
    r.x = oma * s[c].x + ee * k4[c].x;
    r.y = oma * s[c].y + ee * k4[c].y;
    r.z = oma * s[c].z + ee * k4[c].z;
    r.w = oma * s[c].w + ee * k4[c].w;
    *(float4*)(orow + c * 128 + lane * 4) = r;
  }
}

// ---------------------------------------------------------------------------
// meta_in = concat(xsn, memory_context)
// ---------------------------------------------------------------------------
__global__ void concat2(const float* __restrict__ a, const float* __restrict__ c,
                        float* __restrict__ out) {
  const int idx = blockIdx.x * blockDim.x + threadIdx.x;   // over B*1024
  const int bb = idx >> 10;
  const int d = idx & 1023;
  out[idx] = (d < D_) ? a[(size_t)bb * D_ + d] : c[(size_t)bb * D_ + (d - D_)];
}

// ---------------------------------------------------------------------------
// modulated = i_norm * (1 + tanh(gamma)) + beta   (gamma/beta from meta_out)
// ---------------------------------------------------------------------------
__global__ void modulate(const float* __restrict__ inorm, const float* __restrict__ meta_out,
                         float* __restrict__ mod) {
  const int idx = blockIdx.x * blockDim.x + threadIdx.x;   // over B*512
  const int bb = idx >> 9;
  const int d = idx & 511;
  const float* mo = meta_out + (size_t)bb * 1026;
  const float gamma = mo[d];
  const float beta = mo[512 + d];
  mod[idx] = inorm[idx] * (1.0f + tanhf(gamma)) + beta;
}

// ---------------------------------------------------------------------------
// Per-row alpha/eta gates (w_alpha, w_eta are 1xD).
// ---------------------------------------------------------------------------
__global__ void alpha_eta(const float* __restrict__ mod,
                          const float* __restrict__ w_a, const float* __restrict__ b_a,
                          const float* __restrict__ w_e, const float* __restrict__ b_e,
                          const float* __restrict__ meta_out,
                          float* __restrict__ alpha, float* __restrict__ eta) {
  const int bb = blockIdx.x;
  const int t = threadIdx.x;
  const float* m = mod + (size_t)bb * D_;
  const float m0 = m[t], m1 = m[t + 256];
  __shared__ float sa[256], se[256];
  sa[t] = m0 * w_a[t] + m1 * w_a[t + 256];
  se[t] = m0 * w_e[t] + m1 * w_e[t + 256];
  __syncthreads();
#pragma unroll
  for (int off = 128; off > 0; off >>= 1) {
    if (t < off) { sa[t] += sa[t + off]; se[t] += se[t + off]; }
    __syncthreads();
  }
  if (t == 0) {
    const float* mo = meta_out + (size_t)bb * 1026;
    const float za = sa[0] + b_a[0] + mo[1024];
    const float ze = se[0] + b_e[0] + mo[1025];
    alpha[bb] = 1.0f / (1.0f + __expf(-za));
    eta[bb] = (1.0f / (1.0f + __expf(-ze))) * 0.044194173824159216f;  // D^-0.5
  }
}

// ---------------------------------------------------------------------------
extern "C" void kernel_launch(void* const* d_in, const int* in_sizes, int n_in,
                              void* d_out, int out_size, void* d_ws, size_t ws_size,
                              hipStream_t stream) {
  (void)in_sizes; (void)n_in; (void)out_size; (void)ws_size;

  const float* item_emb    = (const float*)d_in[0];
  const float* old_state   = (const float*)d_in[1];
  const float* user_static = (const float*)d_in[2];
  const float* w_q   = (const float*)d_in[3];
  const float* w_k   = (const float*)d_in[4];
  const float* w_v   = (const float*)d_in[5];
  const float* w_a   = (const float*)d_in[6];
  const float* b_a   = (const float*)d_in[7];
  const float* w_e   = (const float*)d_in[8];
  const float* b_e   = (const float*)d_in[9];
  const float* mc_w1 = (const float*)d_in[10];
  const float* mc_b1 = (const float*)d_in[11];
  const float* mc_ln_g = (const float*)d_in[12];
  const float* mc_ln_b = (const float*)d_in[13];
  const float* mc_w2 = (const float*)d_in[14];
  const float* mc_b2 = (const float*)d_in[15];
  const float* n1_g  = (const float*)d_in[16];
  const float* n1_b  = (const float*)d_in[17];

  float* ws = (float*)d_ws;
  float* xsn      = ws;                       // B*D
  float* qbuf     = xsn + B_ * D_;            // B*D
  float* mcbuf    = qbuf + B_ * D_;           // B*D
  float* meta_in  = mcbuf + B_ * D_;          // B*2D
  float* hbuf     = meta_in + B_ * 2 * D_;    // B*D
  float* meta_out = hbuf + B_ * D_;           // B*1026
  float* inorm    = meta_out + B_ * 1026;     // B*D
  float* modbuf   = inorm + B_ * D_;          // B*D
  float* kbuf     = modbuf + B_ * D_;         // B*D
  float* vbuf     = kbuf + B_ * D_;           // B*D
  float* alphab   = vbuf + B_ * D_;           // B
  float* etab     = alphab + B_;              // B

  float* out = (float*)d_out;

  // x_static_norm, i_norm
  ln_rows<<<B_, 256, 0, stream>>>(user_static, n1_g, n1_b, xsn, 0);
  ln_rows<<<B_, 256, 0, stream>>>(item_emb, n1_g, n1_b, inorm, 0);

  // q = l2norm(xsn @ w_q^T)       [256x512x512, WMMA fast path]
  gemm_wmma_f32<<<128, 128, 0, stream>>>(xsn, w_q, nullptr, qbuf, 512, 512, 32, 512);
  l2norm_rows<<<B_, 256, 0, stream>>>(qbuf);

  // memory_context = old_state @ q   (first pass over 268 MB state)
  state_matvec<<<dim3(B_, D_ / 8), 256, 0, stream>>>(old_state, qbuf, mcbuf);

  // meta MLP
  concat2<<<(B_ * 1024) / 256, 256, 0, stream>>>(xsn, mcbuf, meta_in);
  gemm_wmma_f32<<<128, 128, 0, stream>>>(meta_in, mc_w1, mc_b1, hbuf, 1024, 512, 32, 512);
  ln_rows<<<B_, 256, 0, stream>>>(hbuf, mc_ln_g, mc_ln_b, hbuf, 1);
  // meta_out: N=1026 -> 65 N-tiles (padded to 1040), ldc=1026, guarded kernel
  gemm_wmma_f32_pad<<<260, 128, 0, stream>>>(hbuf, mc_w2, mc_b2, meta_out, 1026, 512, 1026, 65, 1040);

  // modulation + k/v/alpha/eta
  modulate<<<(B_ * D_) / 256, 256, 0, stream>>>(inorm, meta_out, modbuf);
  gemm_wmma_f32<<<128, 128, 0, stream>>>(modbuf, w_k, nullptr, kbuf, 512, 512, 32, 512);
  l2norm_rows<<<B_, 256, 0, stream>>>(kbuf);
  gemm_wmma_f32<<<128, 128, 0, stream>>>(modbuf, w_v, nullptr, vbuf, 512, 512, 32, 512);
  alpha_eta<<<B_, 256, 0, stream>>>(modbuf, w_a, b_a, w_e, b_e, meta_out, alphab, etab);

  // fused pred/error/update: single read + single write of state
  state_update<<<dim3(B_, D_ / 8), 256, 0, stream>>>(old_state, kbuf, vbuf, alphab, etab, out);
}